// RoutingLayer_34102040330539
// MI455X (gfx1250) — compile-verified
//
#include <hip/hip_runtime.h>
#include <hip/hip_bf16.h>

// RoutingLayer on MI455X (gfx1250).
// s[b,k,i] = sum_j in[b,j,i] * W[k,j,i] * (1 + bias[k,j])   (softmax over size-1 axis == 1)
// out[b,k,i] = (1 - 1/(exp(||s||)+eps)) * s / (||s||+eps)
// 128 independent GEMMs (M=1024,K=16,N=32), one per feature, via V_WMMA_F32_16X16X4_F32.

#define N_BATCH 1024
#define IN_CH   16
#define NCAPS   32
#define FEAT    128
#define SPITCH  132          // LDS row pitch in floats: 16B-aligned, odd/64 bank stride
#define EPS_F   1e-21f

typedef __attribute__((ext_vector_type(2))) float v2f;
typedef __attribute__((ext_vector_type(8))) float v8f;

__global__ __launch_bounds__(256) void caps_routing_wmma(
    const float* __restrict__ in,    // [1024,16,128]
    const float* __restrict__ W,     // [32,16,128]
    const float* __restrict__ bias,  // [32,16,1]
    float* __restrict__ out)         // [1024,32,128]
{
    extern __shared__ float smem[];
    float* s_lds = smem;                    // 256 rows * SPITCH floats (s tile)
    float* norms = smem + 256 * SPITCH;     // 256 floats (per (m,n) sum of squares)

    const int tid = threadIdx.x;
    const int wv  = tid >> 5;       // wave id 0..7 (owns features wv*16 .. wv*16+15)
    const int ln  = tid & 31;
    const int hi  = ln >> 4;        // lane half (K-pair select in A/B fragments)
    const int lo  = ln & 15;        // M index (A) / N index (B)

    const int bt = (int)blockIdx.x >> 1;    // 64 batch tiles
    const int kt = (int)blockIdx.x & 1;     // 2 cap tiles
    const int b0 = bt * 16;
    const int k0 = kt * 16;

    norms[tid] = 0.0f;
    __syncthreads();

    // Per-lane base pointers: A row = batch (b0+lo), B row = cap (k0+lo).
    const float* Abase = in   + (size_t)(b0 + lo) * (IN_CH * FEAT);
    const float* Bbase = W    + (size_t)(k0 + lo) * (IN_CH * FEAT);
    const float* cbase = bias + (size_t)(k0 + lo) * IN_CH;

    __builtin_prefetch(Abase + wv * 16, 0, 0);   // global_prefetch_b8

    float nrm[8];
#pragma unroll
    for (int r = 0; r < 8; ++r) nrm[r] = 0.0f;

    // 4 chunks of 4 features each (16 features per wave)
    for (int ic = 0; ic < 4; ++ic) {
        const int fi = wv * 16 + ic * 4;    // feature base, multiple of 4

        // Load A/B fragment rows for 4 features at once (float4 per row).
        // WMMA q (K = 4q..4q+3): VGPR v of lane-half 'hi' supplies K = 4q + 2*hi + v.
        float4 Arow[8], Brow[8];
#pragma unroll
        for (int q = 0; q < 4; ++q) {
#pragma unroll
            for (int c = 0; c < 2; ++c) {
                const int j  = 4 * q + 2 * hi + c;
                const int ra = 2 * q + c;
                Arow[ra] = *(const float4*)(Abase + j * FEAT + fi);
                const float cf = 1.0f + cbase[j];        // c = softmax(singleton)+b = 1+b
                float4 wb = *(const float4*)(Bbase + j * FEAT + fi);
                wb.x *= cf; wb.y *= cf; wb.z *= cf; wb.w *= cf;
                Brow[ra] = wb;
            }
        }

#pragma unroll
        for (int t = 0; t < 4; ++t) {
            v8f acc = {};
#pragma unroll
            for (int q = 0; q < 4; ++q) {
                v2f a, b2;
                a[0]  = ((const float*)&Arow[2 * q + 0])[t];
                a[1]  = ((const float*)&Arow[2 * q + 1])[t];
                b2[0] = ((const float*)&Brow[2 * q + 0])[t];
                b2[1] = ((const float*)&Brow[2 * q + 1])[t];
                // D = A(16x4,f32) x B(4x16,f32) + C(16x16,f32)
                acc = __builtin_amdgcn_wmma_f32_16x16x4_f32(
                    /*neg_a=*/false, a, /*neg_b=*/false, b2,
                    /*c_mod=*/(short)0, acc, /*reuse_a=*/false, /*reuse_b=*/false);
            }
            const int i = fi + t;
            // C/D layout: lane 0-15 VGPR r -> M=r; lane 16-31 -> M=8+r; N=lo.
#pragma unroll
            for (int r = 0; r < 8; ++r) {
                const float v = acc[r];
                nrm[r] += v * v;
                s_lds[((r + 8 * hi) * 16 + lo) * SPITCH + i] = v;
            }
        }
    }

    // Cross-wave (cross-feature) norm reduction: ds_add_f32 atomics.
#pragma unroll
    for (int r = 0; r < 8; ++r)
        atomicAdd(&norms[(r + 8 * hi) * 16 + lo], nrm[r]);
    __syncthreads();

    // Squash + coalesced writeback: each wave handles 32 (m,n) rows,
    // lane L writes features 4L..4L+3 -> 512B contiguous per row per wave.
    for (int rr = 0; rr < 32; ++rr) {
        const int R = wv * 32 + rr;
        const int m = R >> 4;
        const int n = R & 15;
        const float n2    = norms[R];
        const float nl    = __builtin_sqrtf(n2);
        const float scale = (1.0f - 1.0f / (__expf(nl) + EPS_F)) / (nl + EPS_F);
        float4 sv = *(const float4*)(&s_lds[R * SPITCH + 4 * ln]);
        sv.x *= scale; sv.y *= scale; sv.z *= scale; sv.w *= scale;
        *(float4*)(out + ((size_t)(b0 + m) * NCAPS + (k0 + n)) * FEAT + 4 * ln) = sv;
    }
}

extern "C" void kernel_launch(void* const* d_in, const int* in_sizes, int n_in,
                              void* d_out, int out_size, void* d_ws, size_t ws_size,
                              hipStream_t stream) {
    const float* in   = (const float*)d_in[0];  // [1024,16,128]
    const float* W    = (const float*)d_in[1];  // [32,16,128]
    const float* bias = (const float*)d_in[2];  // [32,16,1]
    float* out = (float*)d_out;                 // [1024,32,128]

    const size_t shmem = (size_t)(256 * SPITCH + 256) * sizeof(float); // ~133 KB (<=320 KB LDS)
    dim3 grid(128), block(256);
    hipLaunchKernelGGL(caps_routing_wmma, grid, block, shmem, stream, in, W, bias, out);
}